// GATLayer_32770600468656
// MI455X (gfx1250) — compile-verified
//
#include <hip/hip_runtime.h>
#include <hip/hip_bf16.h>

typedef __attribute__((ext_vector_type(2))) float    v2f;
typedef __attribute__((ext_vector_type(8))) float    v8f;
typedef __attribute__((ext_vector_type(4))) unsigned v4u;
typedef __attribute__((ext_vector_type(8))) int      v8i;

#define GAT_SLOPE 0.2f

// ---------------------------------------------------------------------------
// Kernel 1: h = x @ W   (M = #nodes, K = 128, N = 128)
//   - W (64 KB) staged into LDS by the Tensor Data Mover (tensor_load_to_lds)
//   - f32 WMMA 16x16x4, two independent accumulator chains per wave
// ---------------------------------------------------------------------------
__global__ __launch_bounds__(256) void gat_gemm_wmma(
    const float* __restrict__ x, const float* __restrict__ W,
    float* __restrict__ h, int nrows)
{
  __shared__ float Ws[128 * 128];                 // 64 KB of W staged in LDS

  // ---- TDM: DMA the whole 128x128 f32 W tile global -> LDS ----------------
  if (threadIdx.x == 0) {
    const unsigned lds_off = (unsigned)(size_t)(&Ws[0]); // flat LDS addr[31:0]
    const unsigned long long ga = (unsigned long long)(const void*)W;

    v4u g0;
    g0[0] = 1u;                                    // count=1, user descriptor
    g0[1] = lds_off;                               // lds_addr
    g0[2] = (unsigned)(ga & 0xFFFFFFFFu);          // global_addr[31:0]
    g0[3] = (unsigned)((ga >> 32) & 0x01FFFFFFu)   // global_addr[56:32]
          | (2u << 30);                            // type = 2 ("image")

    v8i g1;
    g1[0] = (int)(2u << 16);     // wg_mask=0, data_size=2 (4B), no flags
    g1[1] = (int)(128u << 16);   // atomic_bar_addr=0 | tensor_dim0 = 128
    g1[2] = (int)(128u << 16);   // tensor_dim0 hi=0  | tensor_dim1 = 128
    g1[3] = (int)(128u << 16);   // tensor_dim1 hi=0  | tile_dim0  = 128
    g1[4] = (int)(128u);         // tile_dim1 = 128   | tile_dim2  = 0
    g1[5] = (int)(128u);         // tensor_dim0_stride = 128
    g1[6] = 0;                   // stride0 hi | tensor_dim1_stride lo
    g1[7] = 0;                   // tensor_dim1_stride hi
    asm volatile("tensor_load_to_lds %0, %1" :: "s"(g0), "s"(g1) : "memory");
  }
  __builtin_amdgcn_s_wait_tensorcnt(0);           // issuing wave: DMA done
  __syncthreads();                                 // release all waves

  const int lane = threadIdx.x & 31;
  const int wave = threadIdx.x >> 5;
  const int hl   = lane >> 4;                     // 0: lanes 0-15, 1: 16-31
  const int lm   = lane & 15;

  const int row0 = (blockIdx.x * 8 + wave) * 16;  // this wave's 16-row tile
  int arow = row0 + lm;
  if (arow >= nrows) arow = 0;                    // clamp: EXEC all-1 for WMMA
  const float* xrow = x + (size_t)arow * 128;

  // A fragments for whole K (32 steps of K=4).
  // ISA layout (32-bit A 16x4): lanes 0-15 hold K=0,1; lanes 16-31 hold K=2,3.
  v2f afrag[32];
  #pragma unroll
  for (int ks = 0; ks < 32; ++ks)
    afrag[ks] = *(const v2f*)(xrow + ks * 4 + hl * 2);

  #pragma unroll 1
  for (int nt = 0; nt < 8; nt += 2) {             // two column tiles per pass
    const int col0 = nt * 16 + lm;
    const int col1 = col0 + 16;
    v8f acc0 = {}, acc1 = {};
    #pragma unroll
    for (int ks = 0; ks < 32; ++ks) {
      const int kr = ks * 4 + hl * 2;
      v2f b0, b1;
      b0[0] = Ws[kr * 128 + col0];
      b0[1] = Ws[(kr + 1) * 128 + col0];
      b1[0] = Ws[kr * 128 + col1];
      b1[1] = Ws[(kr + 1) * 128 + col1];
      acc0 = __builtin_amdgcn_wmma_f32_16x16x4_f32(false, afrag[ks], false, b0,
                                                   (short)0, acc0, false, false);
      acc1 = __builtin_amdgcn_wmma_f32_16x16x4_f32(false, afrag[ks], false, b1,
                                                   (short)0, acc1, false, false);
    }
    // C/D layout: VGPR v holds row M=v (lanes 0-15) / M=v+8 (lanes 16-31)
    #pragma unroll
    for (int v = 0; v < 8; ++v) {
      const int r = row0 + v + hl * 8;
      if (r < nrows) {
        h[(size_t)r * 128 + col0] = acc0[v];
        h[(size_t)r * 128 + col1] = acc1[v];
      }
    }
  }
}

// ---------------------------------------------------------------------------
// Kernel 2: per-(node,head) attention scores a_src, a_dst
// ---------------------------------------------------------------------------
__global__ __launch_bounds__(256) void gat_att(
    const float* __restrict__ h, const float* __restrict__ att_s,
    const float* __restrict__ att_d, float* __restrict__ asrc,
    float* __restrict__ adst, int Nn)
{
  int t = blockIdx.x * blockDim.x + threadIdx.x;     // t = node*4 + head
  if (t >= Nn * 4) return;
  int hd = t & 3;
  const float4* hp = (const float4*)(h + (size_t)t * 32);
  const float4* as = (const float4*)(att_s + hd * 32);
  const float4* ad = (const float4*)(att_d + hd * 32);
  float s = 0.f, d = 0.f;
  #pragma unroll
  for (int q = 0; q < 8; ++q) {
    float4 v = hp[q], a = as[q], b = ad[q];
    s += v.x * a.x + v.y * a.y + v.z * a.z + v.w * a.w;
    d += v.x * b.x + v.y * b.y + v.z * b.z + v.w * b.w;
  }
  asrc[t] = s;
  adst[t] = d;
}

// ---------------------------------------------------------------------------
// Kernel 3: init out = bias (broadcast), m = -inf, s = 0
// ---------------------------------------------------------------------------
__global__ __launch_bounds__(256) void gat_init(
    float* __restrict__ out, const float* __restrict__ bias,
    float* __restrict__ mmax, float* __restrict__ ssum, int Nn)
{
  int i = blockIdx.x * blockDim.x + threadIdx.x;
  if (i >= Nn * 128) return;
  out[i] = bias[i & 127];
  if (i < Nn * 4) { mmax[i] = -__builtin_inff(); ssum[i] = 0.f; }
}

// float atomic max via sortable-int trick (exact for non-NaN IEEE floats)
__device__ __forceinline__ void atomicMaxF(float* addr, float val)
{
  if (val >= 0.f) atomicMax((int*)addr, __float_as_int(val));
  else            atomicMin((unsigned int*)addr, __float_as_uint(val));
}

// ---------------------------------------------------------------------------
// Kernel 4: per-edge logits (leaky relu) + segment max. Edges E.. = self loops
// ---------------------------------------------------------------------------
__global__ __launch_bounds__(256) void gat_edge_max(
    const int* __restrict__ ei, const float* __restrict__ asrc,
    const float* __restrict__ adst, float* __restrict__ elog,
    float* __restrict__ mmax, int E, int Nn)
{
  int e = blockIdx.x * blockDim.x + threadIdx.x;
  if (e >= E + Nn) return;
  int s, d;
  if (e < E) { s = ei[e]; d = ei[E + e]; } else { s = d = e - E; }
  #pragma unroll
  for (int hd = 0; hd < 4; ++hd) {
    float v = asrc[s * 4 + hd] + adst[d * 4 + hd];
    v = v > 0.f ? v : v * GAT_SLOPE;
    elog[(size_t)e * 4 + hd] = v;
    atomicMaxF(&mmax[d * 4 + hd], v);
  }
}

// ---------------------------------------------------------------------------
// Kernel 5: p = exp(e - m[dst]); segment sum
// ---------------------------------------------------------------------------
__global__ __launch_bounds__(256) void gat_edge_exp(
    const int* __restrict__ ei, float* __restrict__ elog,
    const float* __restrict__ mmax, float* __restrict__ ssum, int E, int Nn)
{
  int t = blockIdx.x * blockDim.x + threadIdx.x;     // t = edge*4 + head
  if (t >= (E + Nn) * 4) return;
  int e = t >> 2, hd = t & 3;
  int d = (e < E) ? ei[E + e] : (e - E);
  float p = __expf(elog[t] - mmax[d * 4 + hd]);
  elog[t] = p;                                       // reuse scratch: e -> p
  atomicAdd(&ssum[d * 4 + hd], p);
}

// ---------------------------------------------------------------------------
// Kernel 6: one wave per edge: out[dst] += alpha * h[src]   (float4 per lane)
// ---------------------------------------------------------------------------
__global__ __launch_bounds__(256) void gat_edge_scatter(
    const int* __restrict__ ei, const float* __restrict__ p,
    const float* __restrict__ ssum, const float* __restrict__ h,
    float* __restrict__ out, int E, int Nn)
{
  int e = (blockIdx.x * 256 + threadIdx.x) >> 5;     // wave id == edge id
  int lane = threadIdx.x & 31;
  if (e >= E + Nn) return;
  int s, d;
  if (e < E) { s = ei[e]; d = ei[E + e]; } else { s = d = e - E; }
  int hd = lane >> 3;                                // 4 chans/lane, same head
  float alpha = p[(size_t)e * 4 + hd] / (ssum[d * 4 + hd] + 1e-16f);
  const float4 hv = *(const float4*)(h + (size_t)s * 128 + lane * 4);
  float* op = out + (size_t)d * 128 + lane * 4;
  atomicAdd(op + 0, hv.x * alpha);
  atomicAdd(op + 1, hv.y * alpha);
  atomicAdd(op + 2, hv.z * alpha);
  atomicAdd(op + 3, hv.w * alpha);
}

// ---------------------------------------------------------------------------
extern "C" void kernel_launch(void* const* d_in, const int* in_sizes, int n_in,
                              void* d_out, int out_size, void* d_ws, size_t ws_size,
                              hipStream_t stream)
{
  const float* x       = (const float*)d_in[0];
  const int*   ei      = (const int*)  d_in[1];
  const float* W       = (const float*)d_in[2];
  const float* att_src = (const float*)d_in[3];
  const float* att_dst = (const float*)d_in[4];
  const float* bias    = (const float*)d_in[5];
  float* out = (float*)d_out;

  const int Nn = in_sizes[0] / 128;
  const int E  = in_sizes[1] / 2;
  const int ET = E + Nn;

  // workspace layout (floats)
  float* h    = (float*)d_ws;                        // Nn*128
  float* asrc = h    + (size_t)Nn * 128;             // Nn*4
  float* adst = asrc + (size_t)Nn * 4;               // Nn*4
  float* mmax = adst + (size_t)Nn * 4;               // Nn*4
  float* ssum = mmax + (size_t)Nn * 4;               // Nn*4
  float* elog = ssum + (size_t)Nn * 4;               // ET*4

  const int mt = (Nn + 15) / 16;                     // 16-row tiles
  gat_gemm_wmma<<<(mt + 7) / 8, 256, 0, stream>>>(x, W, h, Nn);
  gat_att<<<(Nn * 4 + 255) / 256, 256, 0, stream>>>(h, att_src, att_dst,
                                                    asrc, adst, Nn);
  gat_init<<<(Nn * 128 + 255) / 256, 256, 0, stream>>>(out, bias, mmax, ssum, Nn);
  gat_edge_max<<<(ET + 255) / 256, 256, 0, stream>>>(ei, asrc, adst, elog,
                                                     mmax, E, Nn);
  gat_edge_exp<<<(ET * 4 + 255) / 256, 256, 0, stream>>>(ei, elog, mmax,
                                                         ssum, E, Nn);
  gat_edge_scatter<<<(ET + 7) / 8, 256, 0, stream>>>(ei, elog, ssum, h,
                                                     out, E, Nn);
}